// OneUnkMaterialSparseStack_N_79061757984794
// MI455X (gfx1250) — compile-verified
//
#include <hip/hip_runtime.h>
#include <math.h>

// ---------------------------------------------------------------------------
// Complex helpers (float2 = re, im)
// ---------------------------------------------------------------------------
typedef float2 cf;
typedef __attribute__((ext_vector_type(2))) float v2f;
typedef __attribute__((ext_vector_type(8))) float v8f;

__device__ __forceinline__ cf cmk(float r, float i){ return make_float2(r, i); }
__device__ __forceinline__ cf cmul(cf a, cf b){ return cmk(a.x*b.x - a.y*b.y, a.x*b.y + a.y*b.x); }
__device__ __forceinline__ cf cadd(cf a, cf b){ return cmk(a.x + b.x, a.y + b.y); }
__device__ __forceinline__ cf csub(cf a, cf b){ return cmk(a.x - b.x, a.y - b.y); }
__device__ __forceinline__ cf cscale(cf a, float s){ return cmk(a.x*s, a.y*s); }
__device__ __forceinline__ cf cdiv(cf a, cf b){
    float d = 1.0f / (b.x*b.x + b.y*b.y);
    return cmk((a.x*b.x + a.y*b.y)*d, (a.y*b.x - a.x*b.y)*d);
}
// principal-branch complex sqrt (matches jnp.sqrt on complex64)
__device__ __forceinline__ cf csqrtc(cf z){
    float m = sqrtf(z.x*z.x + z.y*z.y);
    if (z.x >= 0.0f){
        float t  = sqrtf(0.5f*(m + z.x));
        float im = (t > 0.0f) ? 0.5f*z.y/t : 0.0f;
        return cmk(t, im);
    } else {
        float t  = sqrtf(0.5f*(m - z.x));
        float re = (t > 0.0f) ? 0.5f*fabsf(z.y)/t : 0.0f;
        return cmk(re, copysignf(t, z.y));
    }
}
// 5-way register select (materials 0..4) -> v_cndmask chain, no scratch
__device__ __forceinline__ cf pick5(cf a0, cf a1, cf a2, cf a3, cf a4, int m){
    cf r = a0;
    r = (m == 1) ? a1 : r;
    r = (m == 2) ? a2 : r;
    r = (m == 3) ? a3 : r;
    r = (m == 4) ? a4 : r;
    return r;
}

// ---------------------------------------------------------------------------
// Kernel 1: build the [5*SEG, 16] material table in the workspace.
// Segment g (rows g*SEG .. g*SEG+P-1) holds material g's samples:
//   g<4 : col 2g = n_g[p], col 2g+1 = k_g[p]          (from fixed_data)
//   g==4: col 8  = refractive_index[p], col 9 = 0     (dynamic material)
// All other entries are 0 (zero padding rows let the banded GEMM over-read).
// ---------------------------------------------------------------------------
__global__ void build_table_kernel(const float* __restrict__ fixed_data, // [4,3,P]
                                   const float* __restrict__ refr,       // [P]
                                   float* __restrict__ T,                // [5*SEG,16]
                                   int P, int SEG)
{
    int id = blockIdx.x*blockDim.x + threadIdx.x;
    int total = 5*SEG*16;
    if (id >= total) return;
    int r = id >> 4;
    int c = id & 15;
    int g = r / SEG;
    int p = r - g*SEG;
    float v = 0.0f;
    if (p < P){
        if (g < 4){
            if (c == 2*g    ) v = fixed_data[(size_t)g*3*P +   P + p];  // n
            if (c == 2*g + 1) v = fixed_data[(size_t)g*3*P + 2*P + p];  // k
        } else {
            if (c == 8) v = refr[p];                                    // n (k=0)
        }
    }
    T[id] = v;
}

// ---------------------------------------------------------------------------
// Kernel 2: banded interpolation GEMM on the WMMA pipe.
// nc[w, c] = sum_p Wt[w,p] * T[p,c] ; each Wt row has 2 nonzeros (1-t, t).
// One wave per 16-wavelength tile; since wavelengths are sorted, the nonzero
// band inside a tile spans <= ~4 grid rows per material grid, so only 1-2
// V_WMMA_F32_16X16X4_F32 issues per grid segment are needed (f32-exact).
// Output: ncf[w*16 + c], cols (2m,2m+1) = (n,k) of material m, cols 10..15 pad.
// ---------------------------------------------------------------------------
__global__ __launch_bounds__(256) void wmma_interp_kernel(
    const float* __restrict__ wavelengths, // [W]
    const float* __restrict__ fixed_data,  // [4,3,P] (row 0 of each = grid)
    const float* __restrict__ dyn_wl,      // [P]
    const float* __restrict__ T,           // [5*SEG,16]
    float* __restrict__ ncf,               // [W,16]
    int P, int SEG, int W)
{
    const int lane = threadIdx.x & 31;
    const int wv   = threadIdx.x >> 5;
    const int tile = blockIdx.x*8 + wv;
    const int m0   = tile*16;

    const int ln15 = lane & 15;      // A: M row, B/D: N column
    const int half = lane >> 4;      // A: K pair select, B: row-within-VGPR
    const int w    = m0 + ln15;
    const float x  = wavelengths[w];

    v8f acc = {};

    for (int g = 0; g < 5; ++g){
        const float* xp = (g < 4) ? (fixed_data + (size_t)g*3*P) : dyn_wl;

        // per-row interpolation stencil: weights w0@idx0, w1@idx1
        int idx0, idx1; float w0, w1;
        if (x <= xp[0])        { idx0 = idx1 = 0;     w0 = 1.0f; w1 = 0.0f; }
        else if (x >= xp[P-1]) { idx0 = idx1 = P - 1; w0 = 1.0f; w1 = 0.0f; }
        else {
            int lo = 0, hi = P - 1;
            while (hi - lo > 1){ int mid = (lo + hi) >> 1; if (xp[mid] <= x) lo = mid; else hi = mid; }
            float t = (x - xp[lo]) / (xp[lo+1] - xp[lo]);
            idx0 = lo; idx1 = lo + 1; w0 = 1.0f - t; w1 = t;
        }

        // band limits for this tile (wavelengths sorted ascending):
        // lane 0 carries row 0 (min), lane 15 carries row 15 (max)
        int kstart = __builtin_amdgcn_readfirstlane(idx0);
        int kend   = __builtin_amdgcn_readlane(idx1, 15);
        int base   = g*SEG;

        for (int k0 = kstart; k0 <= kend; k0 += 4){
            // A fragment (16x4): lanes 0-15 -> K = k0,k0+1 ; lanes 16-31 -> K = k0+2,k0+3
            int ka = k0 + half*2;
            v2f a;
            a.x = ((ka   == idx0) ? w0 : 0.0f) + ((ka   == idx1) ? w1 : 0.0f);
            a.y = ((ka+1 == idx0) ? w0 : 0.0f) + ((ka+1 == idx1) ? w1 : 0.0f);
            // B fragment (4x16): VGPR0 rows k0,k0+1 (lane halves), VGPR1 rows k0+2,k0+3
            v2f b;
            b.x = T[(size_t)(base + k0     + half)*16 + ln15];
            b.y = T[(size_t)(base + k0 + 2 + half)*16 + ln15];
            acc = __builtin_amdgcn_wmma_f32_16x16x4_f32(
                      false, a, false, b, (short)0, acc, false, false);
        }
    }

    // D (16x16 f32): VGPR r -> M = r + 8*half, lane -> N = ln15
    #pragma unroll
    for (int r = 0; r < 8; ++r){
        int m = r + half*8;
        ncf[(size_t)(m0 + m)*16 + ln15] = acc[r];
    }
}

// ---------------------------------------------------------------------------
// Kernel 3: coherent TMM, s-pol. One lane per (angle, wavelength) point.
// Block = (32 wavelengths, 8 angles) = 256 threads = 8 wave32s.
// Per-wavelength dispersion tile (32 x 64B) staged to LDS with CDNA5
// global_load_async_to_lds_b128 (128 chunks) + s_wait_asynccnt.
// Scalar-invariance: R = |M10/M00|^2, so all 1/t normalizations are skipped.
// ---------------------------------------------------------------------------
#define WTILE 32
#define ATILE 8

__global__ __launch_bounds__(256) void tmm_spol_kernel(
    const float* __restrict__ ncf,        // [W,16] floats
    const float* __restrict__ wavelengths,// [W]
    const float* __restrict__ angles,     // [A]
    const int*   __restrict__ matdist,    // [L+2]
    const float* __restrict__ th_above,   // [nA]
    const float* __restrict__ th_below,   // [nB]
    const float* __restrict__ unk_nm,     // [1] (nm)
    const float* __restrict__ comp,       // [1]
    float* __restrict__ out,              // [A,W]
    int W, int A, int nA, int nB)
{
    const int L = nA + 1 + nB;            // 31 layers

    __shared__ __align__(16) float2 sh_nc[WTILE*8];  // 2048 B tile
    __shared__ float sh_thk[64];
    __shared__ int   sh_mat[66];

    const int tx = threadIdx.x;           // wavelength within tile
    const int ty = threadIdx.y;           // angle within tile
    const int flat = ty*WTILE + tx;

    // stage material map and merged thickness vector
    if (flat >= 160 && flat < 160 + L + 2) sh_mat[flat - 160] = matdist[flat - 160];
    if (flat >= 128 && flat < 128 + L){
        int j = flat - 128;
        float t;
        if      (j <  nA) t = th_above[j];
        else if (j == nA) t = unk_nm[0] * 0.001f * comp[0];  // nm -> um, compressed
        else              t = th_below[j - nA - 1];
        sh_thk[j] = t;
    }

    // CDNA5 async global->LDS staging of the 32-wavelength dispersion tile
    {
        const int nchunk = (WTILE*64)/16;     // 128 x 16B chunks
        if (flat < nchunk){
            unsigned lds_off = (unsigned)(size_t)(&sh_nc[0]) + (unsigned)flat*16u;
            unsigned goff    = (unsigned)(blockIdx.x*(WTILE*64)) + (unsigned)flat*16u;
            asm volatile("global_load_async_to_lds_b128 %0, %1, %2 offset:0"
                         :: "v"(lds_off), "v"(goff), "s"(ncf) : "memory");
        }
        asm volatile("s_wait_asynccnt 0" ::: "memory");
    }
    __syncthreads();

    const int w = blockIdx.x*WTILE + tx;
    const int a = blockIdx.y*ATILE + ty;

    const float lam = wavelengths[w];
    const float ang = angles[a];

    // complex refractive index of the 5 distinct materials at this wavelength
    cf nc0 = sh_nc[tx*8 + 0];
    cf nc1 = sh_nc[tx*8 + 1];
    cf nc2 = sh_nc[tx*8 + 2];
    cf nc3 = sh_nc[tx*8 + 3];
    cf nc4 = sh_nc[tx*8 + 4];

    // Snell invariant: sin0 = n_in * sin(theta)
    const float st = sinf(ang);
    cf nIn  = pick5(nc0, nc1, nc2, nc3, nc4, sh_mat[0]);
    cf sin0 = cscale(nIn, st);
    cf s2   = cmul(sin0, sin0);

    // n*cos(theta) per distinct material (only 5 csqrt per point, not 33)
    #define NCOS(nc) cmul((nc), csqrtc(csub(cmk(1.0f, 0.0f), cdiv(s2, cmul((nc),(nc))))))
    cf no0 = NCOS(nc0);
    cf no1 = NCOS(nc1);
    cf no2 = NCOS(nc2);
    cf no3 = NCOS(nc3);
    cf no4 = NCOS(nc4);
    #undef NCOS

    // first interface (medium 0 | medium 1); skip /t0 (scalar-invariant)
    cf ncp = pick5(no0, no1, no2, no3, no4, sh_mat[0]);
    cf ncc = pick5(no0, no1, no2, no3, no4, sh_mat[1]);
    cf r0  = cdiv(csub(ncp, ncc), cadd(ncp, ncc));
    cf M00 = cmk(1.0f, 0.0f), M01 = r0, M10 = r0, M11 = cmk(1.0f, 0.0f);

    const float k0 = 6.28318530717958647692f / lam;

    for (int j = 1; j <= L; ++j){
        cf ncn = pick5(no0, no1, no2, no3, no4, sh_mat[j + 1]);

        // delta = 2*pi/lam * (n cos) * d  for layer j (medium j)
        cf delta = cscale(ncc, k0 * sh_thk[j - 1]);
        // interface j Fresnel r (t skipped: cancels in M10/M00)
        cf r = cdiv(csub(ncc, ncn), cadd(ncc, ncn));

        // ep = exp(-i*delta) = e^{+Im} (cos Re - i sin Re); em = conj-mag pair
        float snd, csd;
        __sincosf(delta.x, &snd, &csd);
        float epm = __expf(delta.y);
        float emm = __expf(-delta.y);
        cf ep = cmk(epm*csd, -epm*snd);
        cf em = cmk(emm*csd,  emm*snd);
        cf epr = cmul(ep, r);
        cf emr = cmul(em, r);

        // M <- M * [[ep, ep*r],[em*r, em]]
        cf t00 = cadd(cmul(M00, ep ), cmul(M01, emr));
        cf t01 = cadd(cmul(M00, epr), cmul(M01, em ));
        cf t10 = cadd(cmul(M10, ep ), cmul(M11, emr));
        cf t11 = cadd(cmul(M10, epr), cmul(M11, em ));
        M00 = t00; M01 = t01; M10 = t10; M11 = t11;

        ncc = ncn;
    }

    cf ramp = cdiv(M10, M00);
    out[(size_t)a*W + w] = ramp.x*ramp.x + ramp.y*ramp.y;   // R = |r_amp|^2
}

// ---------------------------------------------------------------------------
// Host entry
// ---------------------------------------------------------------------------
extern "C" void kernel_launch(void* const* d_in, const int* in_sizes, int n_in,
                              void* d_out, int out_size, void* d_ws, size_t ws_size,
                              hipStream_t stream)
{
    const float* unk   = (const float*)d_in[0];   // [1] thickness (nm)
    const float* refr  = (const float*)d_in[1];   // [P]
    const float* fixed = (const float*)d_in[2];   // [4,3,P]
    const float* comp  = (const float*)d_in[3];   // [1]
    const float* thA   = (const float*)d_in[4];   // [nA]
    const float* thB   = (const float*)d_in[5];   // [nB]
    const float* wls   = (const float*)d_in[6];   // [W]
    const float* dwl   = (const float*)d_in[7];   // [P]
    const float* angs  = (const float*)d_in[8];   // [A]
    const int*   mat   = (const int*)d_in[9];     // [L+2]
    float* out = (float*)d_out;

    const int P  = in_sizes[1];
    const int W  = in_sizes[6];
    const int A  = in_sizes[8];
    const int nA = in_sizes[4];
    const int nB = in_sizes[5];
    const int SEG = P + 10;                       // padded rows per grid segment

    // workspace: ncf [W,16] f32, then table T [5*SEG,16] f32  (~380 KB total)
    float* ncf = (float*)d_ws;
    float* T   = (float*)d_ws + (size_t)W*16;

    // Pass 1: material table
    int tt = 5*SEG*16;
    build_table_kernel<<<(tt + 255)/256, 256, 0, stream>>>(fixed, refr, T, P, SEG);

    // Pass 2: banded interpolation GEMM on the WMMA pipe (one wave per 16-w tile)
    wmma_interp_kernel<<<(W/16)/8, 256, 0, stream>>>(wls, fixed, dwl, T, ncf, P, SEG, W);

    // Pass 3: TMM (one lane per (angle, wavelength) point)
    dim3 blk(WTILE, ATILE);                       // 256 threads = 8 wave32s
    dim3 grd(W/WTILE, A/ATILE);                   // 4096/32=128, 128/8=16
    tmm_spol_kernel<<<grd, blk, 0, stream>>>(
        ncf, wls, angs, mat, thA, thB, unk, comp, out, W, A, nA, nB);
}